// ErosionLayer_65901978190003
// MI455X (gfx1250) — compile-verified
//
#include <hip/hip_runtime.h>
#include <stdint.h>

#define W       512
#define WMASK   511
#define LOGW    9
#define BATCH   8
#define ITERS   10
#define NCELL   (BATCH * W * W)
#define EPSV    1e-10f
#define E8C     0.00033546262790251185f   /* exp(-8) */
#define CELLW   (200.0f / 512.0f)
#define TWO_PI  6.2831853071795864f

#define TX 32
#define TY 8
// Phase B halo: [-1,+1] in each dim
#define HX (TX + 2)
#define HY (TY + 2)
// Phase A halo: [-1,+2] in each dim (bilinear taps reach x+2 / y+2)
#define AX (TX + 3)
#define AY (TY + 3)

// ---------------------------------------------------------------------------
// CDNA5 async global->LDS copy (ASYNCcnt-tracked), per CDNA5 ISA
// ---------------------------------------------------------------------------
__device__ __forceinline__ void async_g2l_b32(void* lds_dst, const float* gsrc) {
    uint32_t laddr = (uint32_t)(uintptr_t)lds_dst;  // low 32 bits of generic = LDS offset
    asm volatile("global_load_async_to_lds_b32 %0, %1, off"
                 :
                 : "v"(laddr), "v"(gsrc)
                 : "memory");
}
__device__ __forceinline__ void wait_async_zero() {
    asm volatile("s_wait_asynccnt 0" ::: "memory");
}

// soft_floor(x,m) = relu(x-m) + min(e^-8, exp(min(x-m-8,0))) + m
__device__ __forceinline__ float soft_floor(float x, float m) {
    float z = x - m;
    return fmaxf(z, 0.0f) + fminf(E8C, __expf(fminf(z - 8.0f, 0.0f))) + m;
}

// P layout: 0 rain_pow, 1 grav_pow, 2 mhd/cellw, 3 scc_pow, 4 dis_pow,
//           5 dep_pow, 6 maxhd, 7 evap_keep, 8 alpha
__global__ void prep_kernel(const float* alpha, const float* rain_rate,
                            const float* evap, const float* min_hd,
                            const float* grav, const float* scc,
                            const float* dis, const float* dep,
                            const float* max_hd, float* __restrict__ P) {
    if (threadIdx.x == 0 && blockIdx.x == 0) {
        P[0] = exp2f(*rain_rate);
        P[1] = exp2f(*grav);
        P[2] = exp2f(*min_hd) / CELLW;
        P[3] = exp2f(*scc);
        P[4] = exp2f(*dis);
        P[5] = exp2f(*dep);
        P[6] = *max_hd;
        P[7] = 1.0f - exp2f(*evap);
        P[8] = *alpha;
    }
}

__global__ __launch_bounds__(256) void init_kernel(
    const float4* __restrict__ in_t, const float4* __restrict__ orig_t,
    float4* __restrict__ t0, float4* __restrict__ vel,
    float4* __restrict__ sed, float4* __restrict__ wat,
    const float* __restrict__ P) {
    int i = blockIdx.x * 256 + threadIdx.x;
    float a = P[8];
    float4 it = in_t[i], ot = orig_t[i];
    float4 r;
    r.x = 0.5f * (1.0f + (a * it.x + (1.0f - a) * ot.x));
    r.y = 0.5f * (1.0f + (a * it.y + (1.0f - a) * ot.y));
    r.z = 0.5f * (1.0f + (a * it.z + (1.0f - a) * ot.z));
    r.w = 0.5f * (1.0f + (a * it.w + (1.0f - a) * ot.w));
    t0[i] = r;
    float4 zero = make_float4(0.f, 0.f, 0.f, 0.f);
    vel[i] = zero;
    sed[i] = zero;
    wat[i] = zero;
}

// ---------------------------------------------------------------------------
// Phase A: gradient + bilinear sample + velocity/sediment chemistry.
// Terrain halo tile (AY x AX) is fetched into LDS with async loads; all
// stencil and bilinear taps are served from LDS. Wraparound is applied at
// halo-load time; local bilinear indices use the unmasked floor (identical
// to mod-W semantics since |d| <= 1).
// ---------------------------------------------------------------------------
__global__ __launch_bounds__(256) void phaseA_kernel(
    const float* __restrict__ tin, float* __restrict__ tout,
    float* __restrict__ vel,
    const float* __restrict__ sed, float* __restrict__ sedt,
    const float* __restrict__ wat, float* __restrict__ watt,
    float* __restrict__ dcol, float* __restrict__ drow,
    const float* __restrict__ rain, const float* __restrict__ rg,
    const float* __restrict__ P) {
    __shared__ float smT[AY * AX];

    int li = threadIdx.x, lj = threadIdx.y;
    int tid = lj * TX + li;
    int tx0 = blockIdx.x * TX, ty0 = blockIdx.y * TY;
    int bbase = (int)(blockIdx.z << 18);
    const float* t = tin + bbase;

    // async halo fetch: rows [ty0-1, ty0+TY+1], cols [tx0-1, tx0+TX+1]
    for (int c = tid; c < AY * AX; c += 256) {
        int j  = c / AX;
        int i  = c - j * AX;
        int sy = (ty0 - 1 + j) & WMASK;
        int sx = (tx0 - 1 + i) & WMASK;
        async_g2l_b32(&smT[c], t + (sy << LOGW) + sx);
    }
    wait_async_zero();
    __syncthreads();

    int x = tx0 + li, y = ty0 + lj;
    int rem = (y << LOGW) | x;
    int idx = bbase + rem;

    // local center at (lj+1, li+1)
    float tc = smT[(lj + 1) * AX + (li + 1)];
    float tl = smT[(lj + 1) * AX + (li + 0)];
    float tr = smT[(lj + 1) * AX + (li + 2)];
    float tu = smT[(lj + 0) * AX + (li + 1)];
    float td = smT[(lj + 2) * AX + (li + 1)];

    float dxc = 0.5f * (tl - tr);            // ref dx
    float dyc = 0.5f * (tu - td);            // ref dy
    float mag = sqrtf(dxc * dxc + dyc * dyc);
    float r   = rg[rem];
    float d1, d0;                             // d1 = gx (row disp), d0 = gy (col disp)
    if (mag < EPSV) {
        d1 = __cosf(TWO_PI * r);
        d0 = __sinf(TWO_PI * r);
    } else {
        float inv = 1.0f / (mag + EPSV);
        d1 = dxc * inv;
        d0 = dyc * inv;
    }

    // bilinear sample at (col x + d0, row y + d1); taps stay inside the tile
    float cx = (float)x + d0, cy = (float)y + d1;
    float x0f = floorf(cx), y0f = floorf(cy);
    float fx = cx - x0f, fy = cy - y0f;
    int lx0 = (int)x0f - (tx0 - 1);          // in [li, li+2]
    int ly0 = (int)y0f - (ty0 - 1);          // in [lj, lj+2]
    float h00 = smT[ly0 * AX + lx0],     h01 = smT[ly0 * AX + lx0 + 1];
    float h10 = smT[(ly0 + 1) * AX + lx0], h11 = smT[(ly0 + 1) * AX + lx0 + 1];
    float nh  = (h00 * (1.0f - fx) + h01 * fx) * (1.0f - fy)
              + (h10 * (1.0f - fx) + h11 * fx) * fy;
    float hd = tc - nh;

    float rain_pow = P[0], grav_pow = P[1], mhd_cw = P[2], scc_pow = P[3];
    float dis_pow = P[4], dep_pow = P[5], maxhd = P[6];

    float water1 = wat[idx] + fmaxf(rain_pow * rain[rem], 0.0f);
    float v  = vel[idx];
    float vn = sqrtf(soft_floor(v * v + grav_pow * hd, EPSV));
    float new_hd = soft_floor(hd, mhd_cw);
    float cap = new_hd * vn * water1 * scc_pow;

    float s = sed[idx];
    float ftb = (hd < 0.0f) ? 1.0f : 0.0f;
    float min_term = fmaxf(-hd, 0.0f);
    float z1 = min_term - s;
    float first = min_term - (fmaxf(z1, 0.0f) + fminf(E8C, __expf(fminf(z1 - 8.0f, 0.0f))));
    float stb = ((hd - maxhd) > 0.0f) ? 1.0f : 0.0f;
    float second = stb * (maxhd - hd);
    float sdiff = s - cap;
    float third = (1.0f - ftb - stb) *
                  (fmaxf(sdiff * dep_pow, 0.0f) - fmaxf(-sdiff * dis_pow, 0.0f));
    float deposited = soft_floor(first + second + third, -fmaxf(hd, 0.0f));

    sedt[idx] = s - deposited;
    tout[idx] = tc + deposited;
    vel[idx]  = vn;
    watt[idx] = water1;
    dcol[idx] = d0;
    drow[idx] = d1;
}

// ---------------------------------------------------------------------------
// Phase B: displace() as a 9-point gather. Halo tile of {sed,water,d0,d1}
// is pulled into LDS with CDNA5 async global->LDS loads (ASYNCcnt).
// ---------------------------------------------------------------------------
__global__ __launch_bounds__(256) void phaseB_kernel(
    const float* __restrict__ sedt, const float* __restrict__ watt,
    const float* __restrict__ dcol, const float* __restrict__ drow,
    float* __restrict__ sed, float* __restrict__ wat,
    const float* __restrict__ P) {
    __shared__ float4 sm[HY * HX];   // per cell: {sed, water, d0, d1}

    int li = threadIdx.x, lj = threadIdx.y;
    int tid = lj * TX + li;
    int x0 = blockIdx.x * TX, y0 = blockIdx.y * TY;
    int bbase = (int)(blockIdx.z << 18);

    for (int c = tid; c < HY * HX; c += 256) {
        int j  = c / HX;
        int i  = c - j * HX;
        int sy = (y0 - 1 + j) & WMASK;
        int sx = (x0 - 1 + i) & WMASK;
        int g  = bbase + (sy << LOGW) + sx;
        float* lp = (float*)&sm[c];
        async_g2l_b32(lp + 0, sedt + g);
        async_g2l_b32(lp + 1, watt + g);
        async_g2l_b32(lp + 2, dcol + g);
        async_g2l_b32(lp + 3, drow + g);
    }
    wait_async_zero();
    __syncthreads();

    float accs = 0.0f, accw = 0.0f;
#pragma unroll
    for (int oy = 0; oy < 3; ++oy) {
#pragma unroll
        for (int ox = 0; ox < 3; ++ox) {
            float4 v = sm[(lj + oy) * HX + (li + ox)];
            float w0 = fmaxf(0.0f, 1.0f - fabsf(v.z + (float)(ox - 1)));
            float w1 = fmaxf(0.0f, 1.0f - fabsf(v.w + (float)(oy - 1)));
            float w  = w0 * w1;
            accs += v.x * w;
            accw += v.y * w;
        }
    }
    int g = bbase + ((y0 + lj) << LOGW) + (x0 + li);
    sed[g] = accs;
    wat[g] = accw * P[7];   // * (1 - 2^evaporation_rate)
}

__global__ __launch_bounds__(256) void final_kernel(const float4* __restrict__ t,
                                                    float4* __restrict__ out) {
    int i = blockIdx.x * 256 + threadIdx.x;
    float4 v = t[i];
    out[i] = make_float4(2.0f * v.x - 1.0f, 2.0f * v.y - 1.0f,
                         2.0f * v.z - 1.0f, 2.0f * v.w - 1.0f);
}

// ---------------------------------------------------------------------------
extern "C" void kernel_launch(void* const* d_in, const int* in_sizes, int n_in,
                              void* d_out, int out_size, void* d_ws, size_t ws_size,
                              hipStream_t stream) {
    (void)in_sizes; (void)n_in; (void)out_size; (void)ws_size;
    const float* input_terrain = (const float*)d_in[0];
    const float* orig_terrain  = (const float*)d_in[1];
    const float* rain_all      = (const float*)d_in[2];
    const float* rg_all        = (const float*)d_in[3];

    float* ws = (float*)d_ws;
    const size_t N = (size_t)NCELL;
    float* T0   = ws + 0 * N;
    float* T1   = ws + 1 * N;
    float* vel  = ws + 2 * N;
    float* sed  = ws + 3 * N;
    float* sedt = ws + 4 * N;
    float* wat  = ws + 5 * N;
    float* watt = ws + 6 * N;
    float* dc   = ws + 7 * N;
    float* dr   = ws + 8 * N;
    float* P    = ws + 9 * N;

    prep_kernel<<<1, 64, 0, stream>>>(
        (const float*)d_in[4], (const float*)d_in[5], (const float*)d_in[6],
        (const float*)d_in[7], (const float*)d_in[8], (const float*)d_in[9],
        (const float*)d_in[10], (const float*)d_in[11], (const float*)d_in[12], P);

    init_kernel<<<NCELL / 1024, 256, 0, stream>>>(
        (const float4*)input_terrain, (const float4*)orig_terrain,
        (float4*)T0, (float4*)vel, (float4*)sed, (float4*)wat, P);

    float* tin = T0;
    float* tout = T1;
    for (int it = 0; it < ITERS; ++it) {
        phaseA_kernel<<<dim3(W / TX, W / TY, BATCH), dim3(TX, TY), 0, stream>>>(
            tin, tout, vel, sed, sedt, wat, watt, dc, dr,
            rain_all + (size_t)it * W * W, rg_all + (size_t)it * W * W, P);
        phaseB_kernel<<<dim3(W / TX, W / TY, BATCH), dim3(TX, TY), 0, stream>>>(
            sedt, watt, dc, dr, sed, wat, P);
        float* tmp = tin; tin = tout; tout = tmp;
    }

    final_kernel<<<NCELL / 1024, 256, 0, stream>>>((const float4*)tin,
                                                   (float4*)d_out);
}